// QLayer_76304388980736
// MI455X (gfx1250) — compile-verified
//
#include <hip/hip_runtime.h>
#include <math.h>

// ---------------------------------------------------------------------------
// 14-qubit, 6-layer RX+CNOT-ring circuit simulator for MI455X (gfx1250).
//
//  * state (16384 complex64 = 128 KB) fits in one WGP's 320 KB LDS
//    -> 1 workgroup per batch element, whole circuit LDS-resident.
//    HBM traffic ~100 KB total vs ~10.7 GB for an HBM-resident version.
//  * RX gates fused 4 qubits at a time into a 16x16 complex unitary
//    (Kronecker product) -> state reshaped (1024 x 16) -> 16x16 GEMM
//    per 16-fiber tile using V_WMMA_F32_16X16X4_F32 (fp32 matrix path;
//    precision matches the complex64 reference over 84 gates).
//    Complex GEMM = 4 real GEMMs = 16 chained 16x16x4 WMMAs per tile.
//  * CNOT ring is a GF(2)-linear basis permutation (prefix XOR):
//    implemented as an LDS ping-pong gather, zero FLOPs.
//  * readout <Z_q> = sign-weighted probability reduction in LDS.
//
// WMMA register-layout assumptions (from CDNA5 ISA):
//   A (16x4 f32, 2 VGPRs): lane%16 = M, K = j + 2*(lane/16) for VGPR j
//   B (4x16 f32, 2 VGPRs): assumed mirror: row K = j + 2*(lane/16), col = lane%16
//   C/D (16x16 f32, 8 VGPRs): VGPR v, lane l -> (M = v + 8*(l/16), N = l%16)
//   f32 WMMA has no A/B negate (NEG[1:0] reserved) -> keep -Ui in registers.
// ---------------------------------------------------------------------------

#define NQ             14
#define DIM            16384       // 2^14
#define NLAYERS        6
#define THREADS        256         // 8 wave32 waves
#define TILES_PER_WAVE 8           // 64 M-tiles / 8 waves
#define SMEM_BYTES     (4 * DIM * sizeof(float))   // 2 ping-pong complex bufs

typedef float v2f __attribute__((ext_vector_type(2)));
typedef float v8f __attribute__((ext_vector_type(8)));

// address of element k (0..15) of fiber f (0..1023) when the gated nibble
// sits at bit positions [p, p+3] of the 14-bit state index
static __device__ __forceinline__ unsigned fib_addr(unsigned f, unsigned k, int p) {
    unsigned lowmask = (1u << p) - 1u;
    return (f & lowmask) | (k << p) | ((f >> p) << (p + 4));
}

__global__ __launch_bounds__(THREADS)
void qsim_kernel(const float* __restrict__ x,
                 const float* __restrict__ params,
                 float* __restrict__ out)
{
    extern __shared__ float smem[];
    // buffer layout: [buf0_re | buf0_im | buf1_re | buf1_im], each DIM floats.
    // NOTE: pointers are computed arithmetically at each use site; building a
    // local array of LDS pointers triggers an ld.lld static-initializer error.

    const int tid  = threadIdx.x;
    const int lane = tid & 31;
    const int wid  = tid >> 5;
    const int mrow = lane & 15;     // M row / B column for this lane
    const int hi   = lane >> 4;     // half-wave select
    const int b    = blockIdx.x;

    // ---- initial product state: |psi> = kron_q [cos(x_q/2), -i sin(x_q/2)] ----
    float ch[NQ], sh[NQ];
    #pragma unroll
    for (int q = 0; q < NQ; ++q) {
        float h = 0.5f * x[b * NQ + q];
        ch[q] = cosf(h);
        sh[q] = sinf(h);
    }
    {
        float* re0 = smem;
        float* im0 = smem + DIM;
        for (int i = tid; i < DIM; i += THREADS) {
            float mag = 1.0f;
            #pragma unroll
            for (int q = 0; q < NQ; ++q)
                mag *= ((i >> (13 - q)) & 1) ? sh[q] : ch[q];
            int m = __popc((unsigned)i) & 3;          // phase (-i)^popcount
            float r = 0.f, ii = 0.f;
            if      (m == 0) r  =  mag;
            else if (m == 1) ii = -mag;
            else if (m == 2) r  = -mag;
            else             ii =  mag;
            re0[i] = r;
            im0[i] = ii;
        }
    }
    __syncthreads();

    int cur = 0;

    for (int layer = 0; layer < NLAYERS; ++layer) {
        const float* w = params + layer * NQ;

        #pragma unroll
        for (int pass = 0; pass < 4; ++pass) {
            const int pshift = (pass == 0) ? 0 : (pass == 1) ? 4 : (pass == 2) ? 8 : 10;
            // nibble bit t (LSB-first) -> qubit index, -1 = identity padding
            int qb[4];
            if      (pass == 0) { qb[0] = 13; qb[1] = 12; qb[2] = 11; qb[3] = 10; }
            else if (pass == 1) { qb[0] =  9; qb[1] =  8; qb[2] =  7; qb[3] =  6; }
            else if (pass == 2) { qb[0] =  5; qb[1] =  4; qb[2] =  3; qb[3] =  2; }
            else                { qb[0] = -1; qb[1] = -1; qb[2] =  1; qb[3] =  0; }

            float ct[4], st[4];
            #pragma unroll
            for (int t = 0; t < 4; ++t) {
                if (qb[t] >= 0) {
                    float h = 0.5f * w[qb[t]];
                    ct[t] = cosf(h); st[t] = sinf(h);
                } else { ct[t] = 1.f; st[t] = 0.f; }
            }

            // Build U = kron of RX's (symmetric -> B = U^T = U), in B layout.
            // Each entry is purely real or purely imaginary: phase (-i)^m,
            // m = # off-diagonal factors; magnitude = prod of cos/sin.
            v2f Ur[4], Ui[4], nUi[4];
            #pragma unroll
            for (int kc = 0; kc < 4; ++kc) {
                #pragma unroll
                for (int j = 0; j < 2; ++j) {
                    int r = 4 * kc + j + 2 * hi;   // B row (K index)
                    int c = mrow;                  // B col (N index)
                    float mag = 1.f; int m = 0; bool zero = false;
                    #pragma unroll
                    for (int t = 0; t < 4; ++t) {
                        int rt = (r >> t) & 1, cb = (c >> t) & 1;
                        if (qb[t] < 0) { if (rt != cb) zero = true; }
                        else if (rt == cb) mag *= ct[t];
                        else               { mag *= st[t]; ++m; }
                    }
                    float ur = 0.f, ui = 0.f;
                    if (!zero) {
                        if (m & 1) ui = (m == 3) ? mag : -mag;   // (-i)^1=-i, (-i)^3=+i
                        else       ur = (m == 2) ? -mag : mag;   // (-i)^2=-1, else +1
                    }
                    Ur[kc][j] = ur; Ui[kc][j] = ui; nUi[kc][j] = -ui;
                }
            }

            float* re = smem + (unsigned)cur * (2u * DIM);
            float* im = re + DIM;

            for (int t = 0; t < TILES_PER_WAVE; ++t) {
                unsigned fbase = (unsigned)(wid * TILES_PER_WAVE + t) * 16u;
                unsigned f     = fbase + (unsigned)mrow;

                // gather A tiles (Sr, Si) from strided LDS fibers into A layout
                v2f Ar[4], Ai[4];
                #pragma unroll
                for (int kc = 0; kc < 4; ++kc) {
                    unsigned k0 = 4u * kc + 2u * hi;
                    unsigned a0 = fib_addr(f, k0,      pshift);
                    unsigned a1 = fib_addr(f, k0 + 1u, pshift);
                    Ar[kc][0] = re[a0]; Ar[kc][1] = re[a1];
                    Ai[kc][0] = im[a0]; Ai[kc][1] = im[a1];
                }

                // complex GEMM: Dr = Sr*Ur - Si*Ui ; Di = Sr*Ui + Si*Ur
                v8f Dr = {0.f,0.f,0.f,0.f,0.f,0.f,0.f,0.f};
                v8f Di = {0.f,0.f,0.f,0.f,0.f,0.f,0.f,0.f};
                #pragma unroll
                for (int kc = 0; kc < 4; ++kc)
                    Dr = __builtin_amdgcn_wmma_f32_16x16x4_f32(false, Ar[kc], false, Ur[kc],  (short)0, Dr, false, false);
                #pragma unroll
                for (int kc = 0; kc < 4; ++kc)
                    Dr = __builtin_amdgcn_wmma_f32_16x16x4_f32(false, Ai[kc], false, nUi[kc], (short)0, Dr, false, false);
                #pragma unroll
                for (int kc = 0; kc < 4; ++kc)
                    Di = __builtin_amdgcn_wmma_f32_16x16x4_f32(false, Ar[kc], false, Ui[kc],  (short)0, Di, false, false);
                #pragma unroll
                for (int kc = 0; kc < 4; ++kc)
                    Di = __builtin_amdgcn_wmma_f32_16x16x4_f32(false, Ai[kc], false, Ur[kc],  (short)0, Di, false, false);

                // scatter D back in place (fibers are disjoint per wave)
                #pragma unroll
                for (int v = 0; v < 8; ++v) {
                    unsigned ad = fib_addr(fbase + (unsigned)(v + 8 * hi),
                                           (unsigned)mrow, pshift);
                    re[ad] = Dr[v];
                    im[ad] = Di[v];
                }
            }
            __syncthreads();
        }

        // ---- CNOT ring = basis permutation (inverse prefix-XOR gather) ----
        // forward: b'_q = XOR_{j<=q} b_j (q>=1), b'_0 = XOR_{j=1..13} b_j
        // inverse on index word (pos p = 13 - q):
        //   s[0:11] = d ^ (d>>1); s[13] = d13^d0; s[12] = d12^s13
        {
            const float* sre = smem + (unsigned)cur * (2u * DIM);
            const float* sim = sre + DIM;
            float* dre  = smem + (unsigned)(cur ^ 1) * (2u * DIM);
            float* dim_ = dre + DIM;
            for (int d = tid; d < DIM; d += THREADS) {
                unsigned du  = (unsigned)d;
                unsigned s   = (du ^ (du >> 1)) & 0x0FFFu;
                unsigned b13 = ((du >> 13) ^ du) & 1u;
                unsigned b12 = ((du >> 12) & 1u) ^ b13;
                s |= (b13 << 13) | (b12 << 12);
                dre[d]  = sre[s];
                dim_[d] = sim[s];
            }
            cur ^= 1;
            __syncthreads();
        }
    }

    // ---- readout: out[b][q] = sum_i |psi_i|^2 * (1 - 2*bit_q(i)) ----
    const float* fre = smem + (unsigned)cur * (2u * DIM);
    const float* fim = fre + DIM;
    float* red = smem + (unsigned)(cur ^ 1) * (2u * DIM);  // idle buffer scratch
    float acc[NQ];
    #pragma unroll
    for (int q = 0; q < NQ; ++q) acc[q] = 0.f;
    for (int i = tid; i < DIM; i += THREADS) {
        float pr = fre[i], pi = fim[i];
        float p  = pr * pr + pi * pi;
        #pragma unroll
        for (int q = 0; q < NQ; ++q)
            acc[q] += ((i >> (13 - q)) & 1) ? -p : p;
    }
    #pragma unroll
    for (int q = 0; q < NQ; ++q) red[q * THREADS + tid] = acc[q];
    __syncthreads();
    for (int s = THREADS / 2; s > 0; s >>= 1) {
        if (tid < s) {
            #pragma unroll
            for (int q = 0; q < NQ; ++q)
                red[q * THREADS + tid] += red[q * THREADS + tid + s];
        }
        __syncthreads();
    }
    if (tid < NQ) out[b * NQ + tid] = red[tid * THREADS];
}

extern "C" void kernel_launch(void* const* d_in, const int* in_sizes, int n_in,
                              void* d_out, int out_size, void* d_ws, size_t ws_size,
                              hipStream_t stream) {
    const float* x      = (const float*)d_in[0];   // (B, 14)
    const float* params = (const float*)d_in[1];   // (6, 1, 14)
    float* out          = (float*)d_out;           // (B, 14)
    const int B = in_sizes[0] / NQ;                // 512

    (void)hipFuncSetAttribute(reinterpret_cast<const void*>(qsim_kernel),
                              hipFuncAttributeMaxDynamicSharedMemorySize,
                              (int)SMEM_BYTES);

    qsim_kernel<<<dim3(B), dim3(THREADS), SMEM_BYTES, stream>>>(x, params, out);
}